// GCN_57397942944041
// MI455X (gfx1250) — compile-verified
//
#include <hip/hip_runtime.h>
#include <hip/hip_bf16.h>

typedef float v2f __attribute__((ext_vector_type(2)));
typedef float v8f __attribute__((ext_vector_type(8)));

#define INC 32
#define KNB 8

__device__ __forceinline__ float elu1(float v) {
    return v > 0.0f ? v : (__expf(v) - 1.0f);
}

// ---------------------------------------------------------------------------
// Kernel A: conv1 + complex ELU + conv2 (complex GEMM via V_WMMA_F32_16X16X4_F32)
// fused with the two per-row complex dot products the rest of the net needs:
//   tmp[r] = h2[r,:] . Wa^T + ba      (attention scalar)
//   hw[r]  = h2[r,:] . W3^T           (conv3 pre-projection, bias added later)
// One wave handles a 16-row tile. wave32, EXEC all-ones at every WMMA.
// ---------------------------------------------------------------------------
__global__ void k_conv12(const float* __restrict__ xr,  const float* __restrict__ xi,
                         const float* __restrict__ W1r, const float* __restrict__ W1i,
                         const float* __restrict__ b1r, const float* __restrict__ b1i,
                         const float* __restrict__ W2r, const float* __restrict__ W2i,
                         const float* __restrict__ b2r, const float* __restrict__ b2i,
                         const float* __restrict__ War, const float* __restrict__ Wai,
                         const float* __restrict__ bar, const float* __restrict__ bai,
                         const float* __restrict__ W3r, const float* __restrict__ W3i,
                         float2* __restrict__ tmp, float2* __restrict__ hw, int T)
{
    const int lane   = threadIdx.x & 31;
    const int wid    = blockIdx.x * (blockDim.x >> 5) + (threadIdx.x >> 5);
    const int nTiles = (T + 15) >> 4;
    if (wid >= nTiles) return;                 // wave-uniform exit

    const int laneLo = lane & 15;
    const int hiK    = lane >> 4;              // 0: lanes 0-15, 1: lanes 16-31

    // ---- B fragments of W2^T (B[k][n] = W2[n][k]) -------------------------
    // VGPR j of fragment (kk): K = 4*kk + 2*hiK + j, N = nh*16 + laneLo
    v2f Bre[KNB][2], Bim[KNB][2];
#pragma unroll
    for (int kk = 0; kk < 8; ++kk) {
#pragma unroll
        for (int nh = 0; nh < 2; ++nh) {
            const int n = nh * 16 + laneLo;
            const int k = 4 * kk + 2 * hiK;
            Bre[kk][nh] = (v2f){ W2r[n * INC + k], W2r[n * INC + k + 1] };
            Bim[kk][nh] = (v2f){ W2i[n * INC + k], W2i[n * INC + k + 1] };
        }
    }

    // per-output-column constants (column c = nh*16 + laneLo)
    float b2re[2], b2im[2], war[2], wai[2], w3r[2], w3i[2];
#pragma unroll
    for (int nh = 0; nh < 2; ++nh) {
        const int c = nh * 16 + laneLo;
        b2re[nh] = b2r[c];  b2im[nh] = b2i[c];
        war[nh]  = War[c];  wai[nh]  = Wai[c];
        w3r[nh]  = W3r[c];  w3i[nh]  = W3i[c];
    }

    // ---- conv1 + ELU into A fragments -------------------------------------
    const int r  = wid * 16 + laneLo;
    const int rc = min(r, T - 1);              // clamp tail rows (stores guarded)
    const float xre = xr[rc], xim = xi[rc];

    float Are[16], Aim[16];
#pragma unroll
    for (int kk = 0; kk < 8; ++kk) {
#pragma unroll
        for (int j = 0; j < 2; ++j) {
            const int ch = 4 * kk + 2 * hiK + j;
            const float hr = xre * W1r[ch] - xim * W1i[ch] + b1r[ch];
            const float hi = xre * W1i[ch] + xim * W1r[ch] + b1i[ch];
            Are[kk * 2 + j] = elu1(hr);
            Aim[kk * 2 + j] = elu1(hi);
        }
    }

    // ---- complex GEMM: H2 = A * W2^T  (4 real WMMA streams) ---------------
    v8f accRe[2], accIm[2];
#pragma unroll
    for (int nh = 0; nh < 2; ++nh) {
        accRe[nh] = (v8f){0.f,0.f,0.f,0.f,0.f,0.f,0.f,0.f};
        accIm[nh] = (v8f){0.f,0.f,0.f,0.f,0.f,0.f,0.f,0.f};
    }
#pragma unroll
    for (int nh = 0; nh < 2; ++nh) {
#pragma unroll
        for (int kk = 0; kk < 8; ++kk) {
            const v2f aRe  = (v2f){  Are[2 * kk],  Are[2 * kk + 1] };
            const v2f aIm  = (v2f){  Aim[2 * kk],  Aim[2 * kk + 1] };
            const v2f aImN = (v2f){ -Aim[2 * kk], -Aim[2 * kk + 1] };
            // fp32 WMMA: NEG only supports C-negate, so A_im is negated in VALU
            accRe[nh] = __builtin_amdgcn_wmma_f32_16x16x4_f32(
                false, aRe,  false, Bre[kk][nh], (short)0, accRe[nh], false, false);
            accRe[nh] = __builtin_amdgcn_wmma_f32_16x16x4_f32(
                false, aImN, false, Bim[kk][nh], (short)0, accRe[nh], false, false);
            accIm[nh] = __builtin_amdgcn_wmma_f32_16x16x4_f32(
                false, aRe,  false, Bim[kk][nh], (short)0, accIm[nh], false, false);
            accIm[nh] = __builtin_amdgcn_wmma_f32_16x16x4_f32(
                false, aIm,  false, Bre[kk][nh], (short)0, accIm[nh], false, false);
        }
    }

    // ---- fused dual complex dot (Wa, W3) + cross-lane reduction -----------
    // D layout: VGPR v holds row v (lanes 0-15) / row v+8 (lanes 16-31),
    // column = laneLo + 16*nh  -> reduce over 16 lanes with xor 1,2,4,8.
    const float baR = bar[0], baI = bai[0];
#pragma unroll
    for (int v = 0; v < 8; ++v) {
        float tr = 0.f, ti = 0.f, wr = 0.f, wi = 0.f;
#pragma unroll
        for (int nh = 0; nh < 2; ++nh) {
            const float hr = accRe[nh][v] + b2re[nh];
            const float hi = accIm[nh][v] + b2im[nh];
            tr += hr * war[nh] - hi * wai[nh];
            ti += hr * wai[nh] + hi * war[nh];
            wr += hr * w3r[nh] - hi * w3i[nh];
            wi += hr * w3i[nh] + hi * w3r[nh];
        }
#pragma unroll
        for (int m = 1; m < 16; m <<= 1) {
            tr += __shfl_xor(tr, m, 32);
            ti += __shfl_xor(ti, m, 32);
            wr += __shfl_xor(wr, m, 32);
            wi += __shfl_xor(wi, m, 32);
        }
        const int rowOut = wid * 16 + v + 8 * hiK;
        if (laneLo == 0 && rowOut < T) {
            tmp[rowOut] = make_float2(tr + baR, ti + baI);
            hw[rowOut]  = make_float2(wr, wi);
        }
    }
}

// ---------------------------------------------------------------------------
// Kernel B: relu-attention + linearized neighbor aggregation + conv3 bias.
//   i <  M : y[i] = b3 + sum_k (att_k / (sum att + 1e-3)) * hw[idx_k]
//   i >= M : y[i] = b3 + hw[i]        (outlier rows pass through)
// Gathers hit two 4 MB tables -> L2-resident on MI455X (192 MB L2).
// ---------------------------------------------------------------------------
__global__ void k_att(const int* __restrict__ Nneg,
                      const float2* __restrict__ tmp, const float2* __restrict__ hw,
                      const float* __restrict__ b3r, const float* __restrict__ b3i,
                      float2* __restrict__ y, int M, int T)
{
    const int i = blockIdx.x * blockDim.x + threadIdx.x;
    if (i >= T) return;
    const float b3R = b3r[0], b3I = b3i[0];

    if (i < M) {
        const int i0 = Nneg[i];
        const float2 w1 = tmp[i0];
        int   idx[KNB];
        float att[KNB];
        float s = 0.f;
#pragma unroll
        for (int k = 0; k < KNB; ++k) {
            idx[k] = Nneg[(k + 1) * M + i];
            const float2 t2 = tmp[idx[k]];
            float a = w1.x * t2.x + w1.y * t2.y;
            a = a > 0.f ? a : 0.f;
            att[k] = a;
            s += a;
        }
        const float inv = 1.0f / (s + 0.001f);
        float yr = b3R, yi = b3I;
#pragma unroll
        for (int k = 0; k < KNB; ++k) {
            const float L = att[k] * inv;
            const float2 hv = hw[idx[k]];
            yr += L * hv.x;
            yi += L * hv.y;
        }
        y[i] = make_float2(yr, yi);
    } else {
        const float2 hv = hw[i];
        y[i] = make_float2(hv.x + b3R, hv.y + b3I);
    }
}

// ---------------------------------------------------------------------------
// Kernel C: fused final gather with Hermitian mirror.
//   full[q] = q<T ? y[mask_ind_neg[q]] : q==T ? aver : conj(y[mask_ind_neg[2T-q]])
//   out[p]  = full[mask_ind[p]]   (complex64 interleaved)
// ---------------------------------------------------------------------------
__global__ void k_out(const int* __restrict__ mask_ind, const int* __restrict__ mask_ind_neg,
                      const float2* __restrict__ y,
                      const float* __restrict__ avr, const float* __restrict__ avi,
                      float* __restrict__ out, int T, int MN)
{
    const int p = blockIdx.x * blockDim.x + threadIdx.x;
    if (p >= MN) return;
    const int q = mask_ind[p];
    float re, im;
    if (q < T) {
        const float2 v = y[mask_ind_neg[q]];
        re = v.x; im = v.y;
    } else if (q == T) {
        re = avr[0]; im = avi[0];
    } else {
        const float2 v = y[mask_ind_neg[2 * T - q]];
        re = v.x; im = -v.y;
    }
    out[2 * p]     = re;
    out[2 * p + 1] = im;
}

// ---------------------------------------------------------------------------
extern "C" void kernel_launch(void* const* d_in, const int* in_sizes, int n_in,
                              void* d_out, int out_size, void* d_ws, size_t ws_size,
                              hipStream_t stream) {
    const float* xr  = (const float*)d_in[0];
    const float* xi  = (const float*)d_in[1];
    const float* W1r = (const float*)d_in[2];
    const float* W1i = (const float*)d_in[3];
    const float* b1r = (const float*)d_in[4];
    const float* b1i = (const float*)d_in[5];
    const float* W2r = (const float*)d_in[6];
    const float* W2i = (const float*)d_in[7];
    const float* b2r = (const float*)d_in[8];
    const float* b2i = (const float*)d_in[9];
    const float* War = (const float*)d_in[10];
    const float* Wai = (const float*)d_in[11];
    const float* bar = (const float*)d_in[12];
    const float* bai = (const float*)d_in[13];
    const float* W3r = (const float*)d_in[14];
    const float* W3i = (const float*)d_in[15];
    const float* b3r = (const float*)d_in[16];
    const float* b3i = (const float*)d_in[17];
    const float* avr = (const float*)d_in[18];
    const float* avi = (const float*)d_in[19];
    const int* Nneg         = (const int*)d_in[20];
    const int* mask_ind_neg = (const int*)d_in[22];
    const int* mask_ind     = (const int*)d_in[23];

    const int T  = in_sizes[0];            // 501000
    const int M  = in_sizes[20] / (KNB+1); // 500000
    const int MN = in_sizes[23];           // m*n = 1002001

    // workspace: three float2[T] arrays (~12 MB total)
    const size_t seg = (((size_t)T * sizeof(float2)) + 255) & ~(size_t)255;
    char* ws = (char*)d_ws;
    float2* tmp = (float2*)(ws);
    float2* hw  = (float2*)(ws + seg);
    float2* y   = (float2*)(ws + 2 * seg);

    const int nTiles  = (T + 15) / 16;
    const int blocksA = (nTiles + 7) / 8;  // 8 waves (tiles) per 256-thread block

    k_conv12<<<blocksA, 256, 0, stream>>>(xr, xi, W1r, W1i, b1r, b1i,
                                          W2r, W2i, b2r, b2i,
                                          War, Wai, bar, bai, W3r, W3i,
                                          tmp, hw, T);
    k_att<<<(T + 255) / 256, 256, 0, stream>>>(Nneg, tmp, hw, b3r, b3i, y, M, T);
    k_out<<<(MN + 255) / 256, 256, 0, stream>>>(mask_ind, mask_ind_neg, y,
                                                avr, avi, (float*)d_out, T, MN);
}